// Model_37563783971389
// MI455X (gfx1250) — compile-verified
//
#include <hip/hip_runtime.h>

// ---------------------------------------------------------------------------
// GNN (GraphConv x2 + BN + pairwise MLP head) for MI455X (gfx1250, wave32).
// Memory/scatter-bound workload -> keep f32 precision; GEMMs use the native
// f32 tensor op V_WMMA_F32_16X16X4_F32 (one wave computes a 16x64 out tile).
// Weights live in LDS (pre-packed float2 per K-pair -> one ds_load_b64 per
// B operand); A operand is one global_load_b64 per k-step. Operand mode is a
// template parameter so the inner loop is branch-free.
// Scatter-add over 1.2M edges x 64 feats uses global_atomic_add_f32.
// ---------------------------------------------------------------------------

typedef __attribute__((ext_vector_type(2))) float v2f;
typedef __attribute__((ext_vector_type(8))) float v8f;

__device__ __forceinline__ float lrelu01(float t) { return t > 0.0f ? t : 0.01f * t; }

// ---------------------------------------------------------------------------
__global__ void zero_kernel(float* __restrict__ p, int n) {
    int i = blockIdx.x * blockDim.x + threadIdx.x;
    if (i < n) p[i] = 0.0f;
}

__global__ void degree_kernel(const int* __restrict__ src, const int* __restrict__ dst,
                              float* __restrict__ deg_out, float* __restrict__ deg_in, int E) {
    int e = blockIdx.x * blockDim.x + threadIdx.x;
    if (e < E) {
        atomicAdd(&deg_out[src[e]], 1.0f);
        atomicAdd(&deg_in[dst[e]], 1.0f);
    }
}

__global__ void rsqrt_deg_kernel(float* __restrict__ deg, int n) {
    int i = blockIdx.x * blockDim.x + threadIdx.x;
    if (i < n) deg[i] = rsqrtf(fmaxf(deg[i], 1.0f));
}

// ---------------------------------------------------------------------------
// WMMA f32 GEMM: out[M x 64] = Aop[M x K] @ W[K x 64]  (+bias, act epilogue)
// MODE 0: Aop[m,k] = A[m*K+k]
// MODE 1: Aop[m,k] = A[m*K+k] * row_scale[m]
// MODE 2: Aop[m,k] = A[idx0[m]*64+k] - A[idx1[m]*64+k]      (K == 64)
// One wave per 16-row block; 8 waves (256 thr) per workgroup -> 128 rows.
// WMMA f32 16x16x4 operand layout (wave32):
//   A (16x4): all lanes hold M = lane&15; the VGPR pair holds K={0,1} for
//             lanes 0-15 and K={2,3} for lanes 16-31.
//   B (4x16): mirrored (N = lane&15, same K split).
//   C/D (16x16): VGPR r -> M = r + 8*(lane>=16), N = lane&15.
// W is staged in LDS as float2 pairs: lw[(k/2)*64 + c] = {W[k,c], W[k+1,c]}.
// ---------------------------------------------------------------------------
template <int MODE>
__global__ void wmma_gemm64_kernel(const float* __restrict__ A, int K, int M,
                                   const float* __restrict__ W,
                                   float* __restrict__ out,
                                   const float* __restrict__ row_scale,
                                   const int* __restrict__ idx0,
                                   const int* __restrict__ idx1,
                                   const float* __restrict__ bias,
                                   int act /*0 none, 1 leaky-relu 0.01*/) {
    __shared__ v2f lw[64 * 64];   // up to K=128: (K/2)*64 pairs = 32 KB

    // ---- stage W into LDS, packed as (W[2p,c], W[2p+1,c]) ----------------
    for (int i = threadIdx.x; i < (K >> 1) * 64; i += blockDim.x) {
        const int p = i >> 6, c = i & 63;
        v2f w;
        w.x = W[(size_t)(2 * p) * 64 + c];
        w.y = W[(size_t)(2 * p + 1) * 64 + c];
        lw[i] = w;
    }
    __syncthreads();

    const int lane = threadIdx.x & 31;
    const int wave = threadIdx.x >> 5;
    const int lm   = lane & 15;
    const int half = lane >> 4;

    const int row_base = (blockIdx.x * (blockDim.x >> 5) + wave) * 16;
    int arow = row_base + lm;
    if (arow > M - 1) arow = M - 1;           // clamp loads; stores are guarded

    const float* __restrict__ Arow0;
    const float* __restrict__ Arow1 = nullptr;
    float rs = 1.0f;
    if (MODE == 2) {
        Arow0 = A + (size_t)idx0[arow] * 64;
        Arow1 = A + (size_t)idx1[arow] * 64;
        __builtin_prefetch(Arow1, 0, 3);      // global_prefetch_b8
    } else {
        Arow0 = A + (size_t)arow * (size_t)K;
        if (MODE == 1) rs = row_scale[arow];
    }
    __builtin_prefetch(Arow0, 0, 3);

    v8f acc[4];
    acc[0] = v8f{}; acc[1] = v8f{}; acc[2] = v8f{}; acc[3] = v8f{};

    for (int kk = 0; kk < K; kk += 4) {
        const int k0 = kk + 2 * half;         // even -> 8B-aligned v2f loads
        v2f a;
        if (MODE == 2) {
            const v2f a0 = *(const v2f*)(Arow0 + k0);
            const v2f a1 = *(const v2f*)(Arow1 + k0);
            a = a0 - a1;
        } else {
            a = *(const v2f*)(Arow0 + k0);
            if (MODE == 1) { a.x *= rs; a.y *= rs; }
        }
        const int wb = (k0 >> 1) * 64 + lm;   // pair index (k0/2), column lm
        const v2f b0 = lw[wb];
        const v2f b1 = lw[wb + 16];
        const v2f b2 = lw[wb + 32];
        const v2f b3 = lw[wb + 48];
        acc[0] = __builtin_amdgcn_wmma_f32_16x16x4_f32(false, a, false, b0, (short)0, acc[0], false, false);
        acc[1] = __builtin_amdgcn_wmma_f32_16x16x4_f32(false, a, false, b1, (short)0, acc[1], false, false);
        acc[2] = __builtin_amdgcn_wmma_f32_16x16x4_f32(false, a, false, b2, (short)0, acc[2], false, false);
        acc[3] = __builtin_amdgcn_wmma_f32_16x16x4_f32(false, a, false, b3, (short)0, acc[3], false, false);
    }

    const int m_base = row_base + 8 * half;
#pragma unroll
    for (int r = 0; r < 8; ++r) {
        const int m = m_base + r;
        if (m < M) {
#pragma unroll
            for (int nt = 0; nt < 4; ++nt) {
                const int col = nt * 16 + lm;
                float t = acc[nt][r];
                if (bias) t += bias[col];
                if (act == 1) t = lrelu01(t);
                out[(size_t)m * 64 + col] = t;
            }
        }
    }
}

// ---------------------------------------------------------------------------
// Edge scatter-add: agg[dst[e], :] += H[src[e], :]  (64 f32 features/edge).
// Thread t: edge e = t/16, features f = (t%16)*4 -> one float4 gather,
// four global_atomic_add_f32.
// ---------------------------------------------------------------------------
__global__ void scatter_kernel(const float* __restrict__ H,
                               const int* __restrict__ src, const int* __restrict__ dst,
                               float* __restrict__ agg, int E) {
    int gid = blockIdx.x * blockDim.x + threadIdx.x;
    int e = gid >> 4;
    if (e < E) {
        int f = (gid & 15) << 2;
        const float4 h = *(const float4*)(H + (size_t)src[e] * 64 + f);
        float* a = agg + (size_t)dst[e] * 64 + f;
        atomicAdd(a + 0, h.x);
        atomicAdd(a + 1, h.y);
        atomicAdd(a + 2, h.z);
        atomicAdd(a + 3, h.w);
    }
}

// ---------------------------------------------------------------------------
// BatchNorm batch statistics over rows (64 columns), optionally fusing the
// post-aggregation transform t = in * deg_in_rsqrt[n] + bias[c] (mode 0).
// LDS ds_add_f32 partial reduction, then global f32 atomics.
// ---------------------------------------------------------------------------
__global__ void stats_kernel(float* __restrict__ data, const float* __restrict__ rscale,
                             const float* __restrict__ bias, int n_rows, int mode,
                             float* __restrict__ sum, float* __restrict__ sumsq) {
    __shared__ float ls[64];
    __shared__ float lq[64];
    if (threadIdx.x < 64) { ls[threadIdx.x] = 0.0f; lq[threadIdx.x] = 0.0f; }
    __syncthreads();
    const int c = threadIdx.x & 63;
    const int rpb = blockDim.x >> 6;
    float s = 0.0f, q = 0.0f;
    for (int n = blockIdx.x * rpb + (threadIdx.x >> 6); n < n_rows; n += gridDim.x * rpb) {
        float t = data[(size_t)n * 64 + c];
        if (mode == 0) {
            t = t * rscale[n] + bias[c];
            data[(size_t)n * 64 + c] = t;
        }
        s += t;
        q += t * t;
    }
    atomicAdd(&ls[c], s);
    atomicAdd(&lq[c], q);
    __syncthreads();
    if (threadIdx.x < 64) {
        atomicAdd(&sum[threadIdx.x], ls[threadIdx.x]);
        atomicAdd(&sumsq[threadIdx.x], lq[threadIdx.x]);
    }
}

__global__ void bn_finalize_kernel(const float* __restrict__ sum, const float* __restrict__ sumsq,
                                   const float* __restrict__ g, const float* __restrict__ b,
                                   float* __restrict__ scale, float* __restrict__ shift,
                                   float inv_n) {
    int c = threadIdx.x;
    if (c < 64) {
        float mean = sum[c] * inv_n;
        float var  = sumsq[c] * inv_n - mean * mean;
        float sc   = g[c] * rsqrtf(var + 1e-5f);
        scale[c] = sc;
        shift[c] = b[c] - mean * sc;
    }
}

// u = lrelu(v*scale+shift) * deg_out_rsqrt[n]   (BN1 apply + conv2 pre-scale)
__global__ void bn_lrelu_scale_kernel(const float* __restrict__ in, float* __restrict__ out,
                                      const float* __restrict__ scale, const float* __restrict__ shift,
                                      const float* __restrict__ dor, int count) {
    int i = blockIdx.x * blockDim.x + threadIdx.x;
    if (i < count) {
        int n = i >> 6, c = i & 63;
        float t = in[i] * scale[c] + shift[c];
        out[i] = lrelu01(t) * dor[n];
    }
}

// v2 = relu(agg * deg_in_rsqrt[n] + b2)   (in place)
__global__ void bias_relu_kernel(float* __restrict__ d, const float* __restrict__ dir,
                                 const float* __restrict__ bias, int count) {
    int i = blockIdx.x * blockDim.x + threadIdx.x;
    if (i < count) {
        int n = i >> 6, c = i & 63;
        float t = d[i] * dir[n] + bias[c];
        d[i] = fmaxf(t, 0.0f);
    }
}

// z = lrelu(z*scale+shift)   (BN2 apply, in place)
__global__ void bn_lrelu_kernel(float* __restrict__ d, const float* __restrict__ scale,
                                const float* __restrict__ shift, int count) {
    int i = blockIdx.x * blockDim.x + threadIdx.x;
    if (i < count) {
        int c = i & 63;
        d[i] = lrelu01(d[i] * scale[c] + shift[c]);
    }
}

// out[row] = dot(Z[row,:64], fc3_w) + fc3_b
__global__ void fc3_kernel(const float* __restrict__ Z, const float* __restrict__ w,
                           const float* __restrict__ b, float* __restrict__ out, int B) {
    int row = blockIdx.x * blockDim.x + threadIdx.x;
    if (row < B) {
        const float4* z  = (const float4*)(Z + (size_t)row * 64);
        const float4* wv = (const float4*)w;
        float s = 0.0f;
#pragma unroll
        for (int i = 0; i < 16; ++i) {
            float4 a = z[i], c = wv[i];
            s += a.x * c.x + a.y * c.y + a.z * c.z + a.w * c.w;
        }
        out[row] = s + b[0];
    }
}

// ---------------------------------------------------------------------------
extern "C" void kernel_launch(void* const* d_in, const int* in_sizes, int n_in,
                              void* d_out, int out_size, void* d_ws, size_t ws_size,
                              hipStream_t stream) {
    const float* x     = (const float*)d_in[0];
    const int*   src   = (const int*)d_in[1];
    const int*   dst   = (const int*)d_in[2];
    const int*   bidx  = (const int*)d_in[3];
    const float* W1    = (const float*)d_in[4];
    const float* b1    = (const float*)d_in[5];
    const float* W2    = (const float*)d_in[6];
    const float* b2    = (const float*)d_in[7];
    const float* fc1_w = (const float*)d_in[8];
    const float* fc1_b = (const float*)d_in[9];
    const float* fc2_w = (const float*)d_in[10];
    const float* fc2_b = (const float*)d_in[11];
    const float* fc3_w = (const float*)d_in[12];
    const float* fc3_b = (const float*)d_in[13];
    const float* bn1_g = (const float*)d_in[14];
    const float* bn1_b = (const float*)d_in[15];
    const float* bn2_g = (const float*)d_in[16];
    const float* bn2_b = (const float*)d_in[17];

    const int N = in_sizes[0] / 128;       // IDIM = 128
    const int E = in_sizes[1];
    const int B = in_sizes[3] / 2;         // batch_idx is (2, B)
    const int NB = (N > B ? N : B);

    // Workspace layout (floats). Total ~ (2N + 3*NB*64 + 256)*4 B ~= 78 MB.
    float* ws      = (float*)d_ws;
    float* deg_out = ws;                               // N
    float* deg_in  = ws + (size_t)N;                   // N
    float* bufA    = ws + 2 * (size_t)N;               // NB*64  (h1 / h2 / z2)
    float* bufB    = bufA + (size_t)NB * 64;           // NB*64  (agg1,v1pre / agg2,v2)
    float* bufC    = bufB + (size_t)NB * 64;           // NB*64  (u / z1)
    float* sum     = bufC + (size_t)NB * 64;           // 64
    float* sumsq   = sum + 64;                         // 64
    float* bscale  = sum + 128;                        // 64
    float* bshift  = sum + 192;                        // 64

    const int T = 256;
    auto cdiv = [](int a, int b) { return (a + b - 1) / b; };

    const int gemm_blocks_N = cdiv(N, 128);   // 8 waves * 16 rows per block
    const int gemm_blocks_B = cdiv(B, 128);
    const int NF = N * 64;
    const int BF = B * 64;

    // --- degrees ---------------------------------------------------------
    zero_kernel<<<cdiv(2 * N, T), T, 0, stream>>>(deg_out, 2 * N);
    degree_kernel<<<cdiv(E, T), T, 0, stream>>>(src, dst, deg_out, deg_in, E);
    rsqrt_deg_kernel<<<cdiv(2 * N, T), T, 0, stream>>>(deg_out, 2 * N);

    // --- conv1: h1 = (x .* deg_out^-1/2) @ W1 ----------------------------
    wmma_gemm64_kernel<1><<<gemm_blocks_N, T, 0, stream>>>(
        x, 128, N, W1, bufA, deg_out, nullptr, nullptr, nullptr, 0);
    zero_kernel<<<cdiv(NF, T), T, 0, stream>>>(bufB, NF);
    scatter_kernel<<<cdiv(E * 16, T), T, 0, stream>>>(bufA, src, dst, bufB, E);

    // v1pre = agg * deg_in^-1/2 + b1 (fused) ; BN1 batch stats
    zero_kernel<<<1, 128, 0, stream>>>(sum, 128);
    stats_kernel<<<1024, T, 0, stream>>>(bufB, deg_in, b1, N, 0, sum, sumsq);
    bn_finalize_kernel<<<1, 64, 0, stream>>>(sum, sumsq, bn1_g, bn1_b, bscale, bshift, 1.0f / (float)N);
    // u = lrelu(bn1(v1pre)) .* deg_out^-1/2  (conv2 pre-scale fused)
    bn_lrelu_scale_kernel<<<cdiv(NF, T), T, 0, stream>>>(bufB, bufC, bscale, bshift, deg_out, NF);

    // --- conv2: h2 = u @ W2 ---------------------------------------------
    wmma_gemm64_kernel<0><<<gemm_blocks_N, T, 0, stream>>>(
        bufC, 64, N, W2, bufA, nullptr, nullptr, nullptr, nullptr, 0);
    zero_kernel<<<cdiv(NF, T), T, 0, stream>>>(bufB, NF);
    scatter_kernel<<<cdiv(E * 16, T), T, 0, stream>>>(bufA, src, dst, bufB, E);
    // v2 = relu(agg * deg_in^-1/2 + b2)
    bias_relu_kernel<<<cdiv(NF, T), T, 0, stream>>>(bufB, deg_in, b2, NF);

    // --- head: z1 = (v2[i0]-v2[i1]) @ fc1_w + fc1_b  (gather-diff fused) -
    wmma_gemm64_kernel<2><<<gemm_blocks_B, T, 0, stream>>>(
        bufB, 64, B, fc1_w, bufC, nullptr, bidx, bidx + B, fc1_b, 0);
    zero_kernel<<<1, 128, 0, stream>>>(sum, 128);
    stats_kernel<<<1024, T, 0, stream>>>(bufC, nullptr, nullptr, B, 1, sum, sumsq);
    bn_finalize_kernel<<<1, 64, 0, stream>>>(sum, sumsq, bn2_g, bn2_b, bscale, bshift, 1.0f / (float)B);
    bn_lrelu_kernel<<<cdiv(BF, T), T, 0, stream>>>(bufC, bscale, bshift, BF);

    // z2 = lrelu(z1 @ fc2_w + fc2_b)  (bias+act fused in GEMM epilogue)
    wmma_gemm64_kernel<0><<<gemm_blocks_B, T, 0, stream>>>(
        bufC, 64, B, fc2_w, bufA, nullptr, nullptr, nullptr, fc2_b, 1);

    // out = z2 @ fc3_w + fc3_b
    fc3_kernel<<<cdiv(B, T), T, 0, stream>>>(bufA, fc3_w, fc3_b, (float*)d_out, B);
}